// Slice_5317169512990
// MI455X (gfx1250) — compile-verified
//
#include <hip/hip_runtime.h>
#include <hip/hip_bf16.h>
#include <stdint.h>

// Problem constants (fixed by setup_inputs in the reference)
#define NB   2
#define CC   12
#define DD   8
#define HG   16
#define WG   16
#define HH   2048
#define WW   2048
#define GRID_ELEMS (CC * DD * HG * WG)   // 24576 floats = 96 KB per batch

typedef unsigned int v4u __attribute__((ext_vector_type(4)));
typedef int          v8i __attribute__((ext_vector_type(8)));
typedef int          v4i __attribute__((ext_vector_type(4)));

__global__ __launch_bounds__(256) void slice_kernel(
    const float* __restrict__ bgrid,   // [N, C, D, HG, WG]
    const float* __restrict__ guide,   // [N, 1, H, W]
    float* __restrict__ out)           // [N, C, H, W]
{
    __shared__ float s_grid[GRID_ELEMS];   // 96 KB: whole grid for this batch

    const int row = blockIdx.x;            // 0 .. N*H-1
    const int n   = row / HH;
    const int h   = row - n * HH;

    // ---- Stage bilateral grid for batch n into LDS via the Tensor Data Mover ----
    // D# built per CDNA5 ISA ch.8: 1-D tensor, data_size=4B, 24576 elements.
    if (threadIdx.x < 32) {  // wave 0 issues the TDM op once (TDM ignores EXEC)
        const uint64_t gaddr   = (uint64_t)(uintptr_t)(const void*)(bgrid + (size_t)n * GRID_ELEMS);
        const uint32_t lds_off = (uint32_t)(uintptr_t)(void*)&s_grid[0]; // low 32 bits == LDS byte offset

        // Group 0: count=1 | lds_addr | global_addr[56:0] | type=2
        v4u g0 = { 1u,
                   lds_off,
                   (uint32_t)(gaddr & 0xFFFFFFFFu),
                   (uint32_t)((gaddr >> 32) & 0x1FFFFFFu) | 0x80000000u };

        // Group 1: wg_mask=0, data_size=2(4B) | tensor_dim0=24576, tensor_dim1=1,
        //          tile_dim0=24576, tile_dim1=1, tile_dim2=0, dim0_stride=24576
        v8i g1 = { (int)(2u << 16),                         // [31:0]  data_size=4B
                   (int)((GRID_ELEMS & 0xFFFFu) << 16),     // [63:32] barrier_addr=0 | tensor_dim0.lo
                   (int)((GRID_ELEMS >> 16) | (1u << 16)),  // [95:64] tensor_dim0.hi | tensor_dim1.lo=1
                   (int)((GRID_ELEMS & 0xFFFFu) << 16),     // [127:96] tensor_dim1.hi | tile_dim0
                   (int)1,                                  // [159:128] tile_dim1=1 | tile_dim2=0
                   (int)GRID_ELEMS,                         // [191:160] dim0_stride.lo
                   0,                                       // [223:192] dim0_stride.hi | dim1_stride.lo
                   0 };                                     // [255:224] dim1_stride.hi

        v4i g2 = { 1, 1, 0, 0 };   // tensor_dim2=1, tensor_dim3=1, dim2_stride=0, tile_dim3=0
        v4i g3 = { 0, 0, 0, 0 };
        v8i g4 = { 0, 0, 0, 0, 0, 0, 0, 0 };  // extra operand in the 6-arg (clang-23) form

        __builtin_amdgcn_tensor_load_to_lds(g0, g1, g2, g3, g4, 0);
        __builtin_amdgcn_s_wait_tensorcnt(0);
    }
    __syncthreads();

    // ---- Row-uniform x interpolation setup (x coordinate comes from h!) ----
    const float hgc  = (float)h / (float)(HH - 1) * 2.0f - 1.0f;
    const float ix   = (hgc + 1.0f) * 0.5f * (float)(WG - 1);
    const float ix0f = floorf(ix);
    const float fx   = ix - ix0f;
    const int   ix0  = (int)ix0f;
    const int   x0   = min(max(ix0, 0), WG - 1);
    const int   x1   = min(max(ix0 + 1, 0), WG - 1);
    const float vx0  = (ix0 >= 0 && ix0 < WG) ? 1.0f : 0.0f;
    const float vx1  = (ix0 + 1 >= 0 && ix0 + 1 < WG) ? 1.0f : 0.0f;
    const float wx0  = (1.0f - fx) * vx0;
    const float wx1  = fx * vx1;

    const float* gd = guide + ((size_t)n * HH + h) * WW;
    float*       op = out + (size_t)n * CC * HH * WW + (size_t)h * WW;

    for (int w = threadIdx.x; w < WW; w += 256) {
        // Streaming read: non-temporal hint keeps L2 free for the grid.
        const float g = __builtin_nontemporal_load(gd + w);

        // y interpolation (y coordinate comes from w)
        const float wgc  = (float)w / (float)(WW - 1) * 2.0f - 1.0f;
        const float iy   = (wgc + 1.0f) * 0.5f * (float)(HG - 1);
        const float iy0f = floorf(iy);
        const float fy   = iy - iy0f;
        const int   iy0  = (int)iy0f;
        const int   y0   = min(max(iy0, 0), HG - 1);
        const int   y1   = min(max(iy0 + 1, 0), HG - 1);
        const float vy0  = (iy0 >= 0 && iy0 < HG) ? 1.0f : 0.0f;
        const float vy1  = (iy0 + 1 >= 0 && iy0 + 1 < HG) ? 1.0f : 0.0f;
        const float wy0  = (1.0f - fy) * vy0;
        const float wy1  = fy * vy1;

        // z interpolation (z coordinate comes from guide)
        const float iz   = (g + 1.0f) * 0.5f * (float)(DD - 1);
        const float iz0f = floorf(iz);
        const float fz   = iz - iz0f;
        const int   iz0  = (int)iz0f;
        const int   z0   = min(max(iz0, 0), DD - 1);
        const int   z1   = min(max(iz0 + 1, 0), DD - 1);
        const float vz0  = (iz0 >= 0 && iz0 < DD) ? 1.0f : 0.0f;
        const float vz1  = (iz0 + 1 >= 0 && iz0 + 1 < DD) ? 1.0f : 0.0f;
        const float wz0  = (1.0f - fz) * vz0;
        const float wz1  = fz * vz1;

        // 8 tap indices (within one channel slab) + weights, reference order:
        // (z0,y0,x0)(z0,y0,x1)(z0,y1,x0)(z0,y1,x1)(z1,y0,x0)(z1,y0,x1)(z1,y1,x0)(z1,y1,x1)
        const int b00 = z0 * (HG * WG) + y0 * WG;
        const int b01 = z0 * (HG * WG) + y1 * WG;
        const int b10 = z1 * (HG * WG) + y0 * WG;
        const int b11 = z1 * (HG * WG) + y1 * WG;
        const int i0 = b00 + x0, i1 = b00 + x1, i2 = b01 + x0, i3 = b01 + x1;
        const int i4 = b10 + x0, i5 = b10 + x1, i6 = b11 + x0, i7 = b11 + x1;
        const float w0 = wz0 * wy0 * wx0, w1 = wz0 * wy0 * wx1;
        const float w2 = wz0 * wy1 * wx0, w3 = wz0 * wy1 * wx1;
        const float w4 = wz1 * wy0 * wx0, w5 = wz1 * wy0 * wx1;
        const float w6 = wz1 * wy1 * wx0, w7 = wz1 * wy1 * wx1;

        #pragma unroll
        for (int c = 0; c < CC; ++c) {
            const float* sg = s_grid + c * (DD * HG * WG);
            float acc = sg[i0] * w0 + sg[i1] * w1 + sg[i2] * w2 + sg[i3] * w3
                      + sg[i4] * w4 + sg[i5] * w5 + sg[i6] * w6 + sg[i7] * w7;
            // Streaming write-once output: non-temporal store.
            __builtin_nontemporal_store(acc, op + (size_t)c * (HH * (size_t)WW) + w);
        }
    }
}

extern "C" void kernel_launch(void* const* d_in, const int* in_sizes, int n_in,
                              void* d_out, int out_size, void* d_ws, size_t ws_size,
                              hipStream_t stream) {
    const float* bgrid = (const float*)d_in[0];  // [2,12,8,16,16] f32
    const float* guide = (const float*)d_in[1];  // [2,1,2048,2048] f32
    float*       out   = (float*)d_out;          // [2,12,2048,2048] f32
    (void)in_sizes; (void)n_in; (void)out_size; (void)d_ws; (void)ws_size;

    slice_kernel<<<NB * HH, 256, 0, stream>>>(bgrid, guide, out);
}